// BionetworkModel_11098195492961
// MI455X (gfx1250) — compile-verified
//
#include <hip/hip_runtime.h>
#include <hip/hip_bf16.h>
#include <cstdint>

// ---------------------------------------------------------------------------
// Bionetwork sparse recurrent net on MI455X (gfx1250).
//   per iter: s = scatter_add(rec_w * y[cols] -> rows); y = act(s + b_in)
// Edge stream (38.4MB) is L2-resident (192MB L2). We stage it through LDS with
// CDNA5 async global->LDS loads (ASYNCcnt) and scatter with hw f32 atomics.
// Each wave DMAs 256-edge chunks (1KB per array) double-buffered.
// ---------------------------------------------------------------------------

#define LEAK 0.01f
#define ITERS 150

__device__ __forceinline__ float mml_act(float v) {
    v = (v < 0.0f) ? v * LEAK : v;
    return (v > 0.5f) ? (1.0f - 0.25f / v) : v;
}

// ---- setup: b_in = biases, y = 0, s = 0 -----------------------------------
__global__ void init_state_kernel(const float* __restrict__ biases,
                                  float* __restrict__ b_in,
                                  float* __restrict__ y,
                                  float* __restrict__ s, int n) {
    int i = blockIdx.x * blockDim.x + threadIdx.x;
    if (i < n) {
        b_in[i] = biases[i];
        y[i]    = 0.0f;
        s[i]    = 0.0f;
    }
}

// ---- input projection: b_in[idx[j]] = in_w[j]*x[j] + biases[idx[j]] -------
__global__ void input_proj_kernel(const float* __restrict__ x,
                                  const float* __restrict__ in_w,
                                  const int* __restrict__ in_idx,
                                  const float* __restrict__ biases,
                                  float* __restrict__ b_in, int nIn) {
    int j = blockIdx.x * blockDim.x + threadIdx.x;
    if (j < nIn) {
        int idx = in_idx[j];
        b_in[idx] = in_w[j] * x[j] + biases[idx];   // .set semantics (last writer wins)
    }
}

// ---- edge kernel: async-LDS double-buffered streaming SpMV scatter --------
// 256 threads = 8 waves; each wave owns 256-edge chunks (1KB per array),
// staged as 2 x b128 per lane per array (offset applies to LDS+global alike).
__global__ void __launch_bounds__(256)
edge_kernel(const float* __restrict__ w,
            const int* __restrict__ rows,
            const int* __restrict__ cols,
            const float* __restrict__ y,
            float* __restrict__ s, int nEdges) {
    __shared__ __align__(16) float lw[2][8][256];
    __shared__ __align__(16) int   lr[2][8][256];
    __shared__ __align__(16) int   lc[2][8][256];

    const int lane = threadIdx.x & 31;
    const int wid  = threadIdx.x >> 5;
    const int wavesPerBlock = blockDim.x >> 5;
    const int stride  = gridDim.x * wavesPerBlock;
    const int nChunks = nEdges >> 8;                 // 256 edges per chunk

    // stage chunk CH's 3 x 1KB slices into LDS buffer BUF (2 x b128 per lane)
#define STAGE(CH, BUF)                                                          \
    do {                                                                        \
        const int _e0 = (CH) << 8;                                              \
        unsigned long long _aw = (unsigned long long)(uintptr_t)(w    + _e0) +  \
                                 ((unsigned long long)lane << 4);               \
        unsigned long long _ar = (unsigned long long)(uintptr_t)(rows + _e0) +  \
                                 ((unsigned long long)lane << 4);               \
        unsigned long long _ac = (unsigned long long)(uintptr_t)(cols + _e0) +  \
                                 ((unsigned long long)lane << 4);               \
        unsigned _dw = (unsigned)(uintptr_t)&lw[BUF][wid][lane << 2];           \
        unsigned _dr = (unsigned)(uintptr_t)&lr[BUF][wid][lane << 2];           \
        unsigned _dc = (unsigned)(uintptr_t)&lc[BUF][wid][lane << 2];           \
        asm volatile("global_load_async_to_lds_b128 %0, %1, off"                \
                     :: "v"(_dw), "v"(_aw) : "memory");                         \
        asm volatile("global_load_async_to_lds_b128 %0, %1, off offset:512"     \
                     :: "v"(_dw), "v"(_aw) : "memory");                         \
        asm volatile("global_load_async_to_lds_b128 %0, %1, off"                \
                     :: "v"(_dr), "v"(_ar) : "memory");                         \
        asm volatile("global_load_async_to_lds_b128 %0, %1, off offset:512"     \
                     :: "v"(_dr), "v"(_ar) : "memory");                         \
        asm volatile("global_load_async_to_lds_b128 %0, %1, off"                \
                     :: "v"(_dc), "v"(_ac) : "memory");                         \
        asm volatile("global_load_async_to_lds_b128 %0, %1, off offset:512"     \
                     :: "v"(_dc), "v"(_ac) : "memory");                         \
    } while (0)

    int chunk = blockIdx.x * wavesPerBlock + wid;
    int cur = 0;
    if (chunk < nChunks) STAGE(chunk, 0);

    for (; chunk < nChunks; chunk += stride) {
        const int  nxt       = chunk + stride;
        const bool nextValid = nxt < nChunks;
        if (nextValid) {
            // protect LDS WAR: prior buffer's ds_loads must land before DMA overwrite
            asm volatile("s_wait_dscnt 0" ::: "memory");
            STAGE(nxt, cur ^ 1);
            asm volatile("s_wait_asynccnt 6" ::: "memory");  // current buffer done
        } else {
            asm volatile("s_wait_asynccnt 0" ::: "memory");
        }

        const int e = lane << 2;   // this lane's first element in the chunk
        const float4 wv0 = *reinterpret_cast<const float4*>(&lw[cur][wid][e]);
        const int4   rv0 = *reinterpret_cast<const int4*>(&lr[cur][wid][e]);
        const int4   cv0 = *reinterpret_cast<const int4*>(&lc[cur][wid][e]);
        const float4 wv1 = *reinterpret_cast<const float4*>(&lw[cur][wid][128 + e]);
        const int4   rv1 = *reinterpret_cast<const int4*>(&lr[cur][wid][128 + e]);
        const int4   cv1 = *reinterpret_cast<const int4*>(&lc[cur][wid][128 + e]);

        // gather y then scatter-add (global_atomic_add_f32, no-return)
        unsafeAtomicAdd(&s[rv0.x], wv0.x * y[cv0.x]);
        unsafeAtomicAdd(&s[rv0.y], wv0.y * y[cv0.y]);
        unsafeAtomicAdd(&s[rv0.z], wv0.z * y[cv0.z]);
        unsafeAtomicAdd(&s[rv0.w], wv0.w * y[cv0.w]);
        unsafeAtomicAdd(&s[rv1.x], wv1.x * y[cv1.x]);
        unsafeAtomicAdd(&s[rv1.y], wv1.y * y[cv1.y]);
        unsafeAtomicAdd(&s[rv1.z], wv1.z * y[cv1.z]);
        unsafeAtomicAdd(&s[rv1.w], wv1.w * y[cv1.w]);

        cur ^= 1;
    }
#undef STAGE

    // tail edges (nEdges % 256) via plain path
    const int gtid     = blockIdx.x * blockDim.x + threadIdx.x;
    const int gthreads = gridDim.x * blockDim.x;
    for (int e = (nChunks << 8) + gtid; e < nEdges; e += gthreads)
        unsafeAtomicAdd(&s[rows[e]], w[e] * y[cols[e]]);
}

// ---- activation: y = act(s + b_in); s = 0 for next iteration --------------
__global__ void act_kernel(float* __restrict__ s,
                           const float* __restrict__ b_in,
                           float* __restrict__ y, int n) {
    int i = blockIdx.x * blockDim.x + threadIdx.x;
    if (i < n) {
        float v = s[i] + b_in[i];
        s[i] = 0.0f;
        y[i] = mml_act(v);
    }
}

// ---- output projection ----------------------------------------------------
__global__ void out_proj_kernel(const float* __restrict__ y,
                                const float* __restrict__ out_w,
                                const int* __restrict__ out_idx,
                                float* __restrict__ out, int nOut) {
    int j = blockIdx.x * blockDim.x + threadIdx.x;
    if (j < nOut)
        out[j] = out_w[j] * y[out_idx[j]];
}

// ---------------------------------------------------------------------------
extern "C" void kernel_launch(void* const* d_in, const int* in_sizes, int n_in,
                              void* d_out, int out_size, void* d_ws, size_t ws_size,
                              hipStream_t stream) {
    const float* x       = (const float*)d_in[0];
    const float* in_w    = (const float*)d_in[1];
    const float* rec_w   = (const float*)d_in[2];
    const float* biases  = (const float*)d_in[3];
    const float* out_w   = (const float*)d_in[4];
    const int*   in_idx  = (const int*)d_in[5];
    const int*   erows   = (const int*)d_in[6];
    const int*   ecols   = (const int*)d_in[7];
    const int*   out_idx = (const int*)d_in[8];

    const int nIn    = in_sizes[1];
    const int nEdges = in_sizes[2];
    const int nNodes = in_sizes[3];
    const int nOut   = in_sizes[4];

    // workspace: s | y | b_in  (3 * nNodes floats = 1.2 MB)
    float* s    = (float*)d_ws;
    float* y    = s + nNodes;
    float* b_in = y + nNodes;

    const int nodeBlocks = (nNodes + 255) / 256;
    init_state_kernel<<<nodeBlocks, 256, 0, stream>>>(biases, b_in, y, s, nNodes);
    input_proj_kernel<<<(nIn + 127) / 128, 128, 0, stream>>>(x, in_w, in_idx,
                                                             biases, b_in, nIn);

    const int edgeBlocks = 512;  // 4096 waves; 12500 chunks -> ~3 chunks/wave
    for (int it = 0; it < ITERS; ++it) {
        edge_kernel<<<edgeBlocks, 256, 0, stream>>>(rec_w, erows, ecols, y, s, nEdges);
        act_kernel<<<nodeBlocks, 256, 0, stream>>>(s, b_in, y, nNodes);
    }

    out_proj_kernel<<<1, 64, 0, stream>>>(y, out_w, out_idx, (float*)d_out, nOut);
}